// GWASEncoder_5162550690504
// MI455X (gfx1250) — compile-verified
//
#include <hip/hip_runtime.h>

// CDNA5 / gfx1250, wave32.
typedef __attribute__((ext_vector_type(2))) float v2f;
typedef __attribute__((ext_vector_type(8))) float v8f;

#define NROWS 30000
#define TTOK  64
#define DIM   128
#define FEAT  137   // 128 trait + 8 cat + 1 score
#define KPAD  140   // FEAT + bias column (137) + zero pad (138,139)

// out[n,d] = ( sum_f Wext[d,f] * u[n,f] ) / max(s_w, 1e-8)
//   u[0:128]  = sum_t w_t * trait_embed[token]
//   u[128:136]= sum_t w_t * cat_embed[cat]
//   u[136]    = sum_t w_t * s_t
//   u[137]    = s_w = sum_t w_t          (picks up bias column Wext[:,137]=b)
__global__ __launch_bounds__(256)
void gwas_fused_kernel(const int*   __restrict__ tok,
                       const float* __restrict__ sc,
                       const int*   __restrict__ cat,
                       const float* __restrict__ temb,
                       const float* __restrict__ cemb,
                       const float* __restrict__ Wp,
                       const float* __restrict__ bias,
                       float*       __restrict__ out)
{
    __shared__ float U[16][148];     // row stride 148 floats (16B-aligned, de-conflicted)
    __shared__ float dnm[16];
    __shared__ int   sTok[16 * TTOK];
    __shared__ float sSc [16 * TTOK];
    __shared__ int   sCat[16 * TTOK];

    const int tid  = threadIdx.x;
    const int lane = tid & 31;
    const int wave = tid >> 5;
    const int n0   = blockIdx.x * 16;

    // ---- Stage 0: coalesced staging of ids/scores for this block's 16 rows ----
    for (int i = tid; i < 16 * TTOK; i += 256) {
        const int n = n0 + (i >> 6);
        const int t = i & 63;
        sTok[i] = tok[n * TTOK + t];
        sSc[i]  = sc [n * TTOK + t];
        sCat[i] = cat[n * TTOK + t];
    }
    __syncthreads();

    // ---- Stage A: weighted gather-reduce; each wave owns 2 rows ----
    for (int rr = 0; rr < 2; ++rr) {
        const int r = wave * 2 + rr;
        float ax = 0.f, ay = 0.f, az = 0.f, aw = 0.f;
        float cacc = 0.f, sws = 0.f, wsum = 0.f;
        for (int t = 0; t < TTOK; ++t) {
            const int   idx   = r * TTOK + t;
            const int   token = sTok[idx];
            const float s     = sSc[idx];
            const float w     = (token != 0) ? s : 0.0f;
            const float4 e = *(const float4*)(temb + (size_t)token * DIM + lane * 4);
            ax = fmaf(w, e.x, ax);
            ay = fmaf(w, e.y, ay);
            az = fmaf(w, e.z, az);
            aw = fmaf(w, e.w, aw);
            if (lane < 8) cacc = fmaf(w, cemb[sCat[idx] * 8 + lane], cacc);
            sws  = fmaf(w, s, sws);
            wsum += w;
        }
        U[r][lane * 4 + 0] = ax;
        U[r][lane * 4 + 1] = ay;
        U[r][lane * 4 + 2] = az;
        U[r][lane * 4 + 3] = aw;
        if (lane < 8) U[r][128 + lane] = cacc;
        if (lane == 0) {
            U[r][136] = sws;
            U[r][137] = wsum;    // bias column weight
            U[r][138] = 0.0f;
            U[r][139] = 0.0f;
            dnm[r] = fmaxf(wsum, 1e-8f);
        }
    }
    __syncthreads();

    // ---- Stage B: 16x16 output tile per wave via V_WMMA_F32_16X16X4_F32 ----
    // ISA layouts (wave32):
    //  A 16x4 : lane l -> M = l&15; VGPR j -> K = 2*(l>>4) + j
    //  B 4x16 : lane l -> N = l&15; VGPR j -> K = 2*(l>>4) + j
    //  C/D    : VGPR r, lane l -> M = r + 8*(l>>4), N = l&15
    const int d0   = wave * 16;      // 8 waves x 16 cols = 128
    const int half = lane >> 4;
    const int l15  = lane & 15;
    const int dcol = d0 + l15;

    v8f c = {0.f, 0.f, 0.f, 0.f, 0.f, 0.f, 0.f, 0.f};
    #pragma unroll
    for (int k = 0; k < KPAD; k += 4) {
        const int ks = k + half * 2;
        v2f a, b;
        a.x = U[l15][ks];
        a.y = U[l15][ks + 1];
        // Wext[dcol][k] : k<137 -> proj_w, k==137 -> bias, else 0  (resolved at compile time)
        b.x = (ks     < FEAT) ? Wp[dcol * FEAT + ks]
            : ((ks     == FEAT) ? bias[dcol] : 0.0f);
        b.y = (ks + 1 < FEAT) ? Wp[dcol * FEAT + ks + 1]
            : ((ks + 1 == FEAT) ? bias[dcol] : 0.0f);
        c = __builtin_amdgcn_wmma_f32_16x16x4_f32(
                /*neg_a=*/false, a, /*neg_b=*/false, b,
                /*c_mod=*/(short)0, c, /*reuse_a=*/false, /*reuse_b=*/false);
    }

    #pragma unroll
    for (int r = 0; r < 8; ++r) {
        const int m = r + half * 8;
        out[(size_t)(n0 + m) * DIM + dcol] = c[r] / dnm[m];
    }
}

extern "C" void kernel_launch(void* const* d_in, const int* in_sizes, int n_in,
                              void* d_out, int out_size, void* d_ws, size_t ws_size,
                              hipStream_t stream) {
    (void)in_sizes; (void)n_in; (void)out_size; (void)d_ws; (void)ws_size;
    const int*   tok  = (const int*)  d_in[0];
    const float* sc   = (const float*)d_in[1];
    const int*   cat  = (const int*)  d_in[2];
    const float* temb = (const float*)d_in[3];
    const float* cemb = (const float*)d_in[4];
    const float* Wp   = (const float*)d_in[5];
    const float* bias = (const float*)d_in[6];
    float* out = (float*)d_out;

    dim3 grid(NROWS / 16);   // 1875 blocks, each covers 16 rows exactly
    gwas_fused_kernel<<<grid, 256, 0, stream>>>(tok, sc, cat, temb, cemb, Wp, bias, out);
}